// GPN_20993800143345
// MI455X (gfx1250) — compile-verified
//
#include <hip/hip_runtime.h>

// ---------------------------------------------------------------------------
// GPN forward for MI455X (gfx1250, wave32, WMMA f32 16x16x4).
//
//   deg = scatter_add(weight, col);  dis = deg^-0.5 (0 if deg==0)
//   per layer: agg = x + scatter_add(dis[row]*w*dis[col] * x[row], col)
//              t1  = agg @ W1^T + b1                       (WMMA f32)
//              stats(t1) -> mu,rstd
//              t2  = relu(bn(t1)) @ W2^T + b2              (BN fused in A-load)
//              stats(t2); h = relu(bn(t2))                 (elementwise)
//   heads:     t = x @ lp0^T + lp0_b;  t += h1 @ lp2^T + lp2_b
//              out = relu(t) @ out_W^T + out_b             (Nc = 64)
//
// GEMM: one wave per (16-row M-tile, 16-col N-tile). A-tile and B-tile are
// fully preloaded into registers (32 x v2f each, coalesced b64 loads) so the
// 32-step V_WMMA_F32_16X16X4_F32 chain runs with a single loadcnt wait.
// Two interleaved accumulators break the D->C serial dependency.
// ---------------------------------------------------------------------------

#define NNODES 50000
#define NEDGES 800000
#define DF     128
#define DOUT   64
#define BN_EPS 1e-5f

typedef __attribute__((ext_vector_type(2))) float v2f;
typedef __attribute__((ext_vector_type(8))) float v8f;

// ------------------------------- utility kernels ---------------------------

__global__ void k_zero(float* __restrict__ p, int n) {
  int i = blockIdx.x * blockDim.x + threadIdx.x;
  if (i < n) p[i] = 0.0f;
}

__global__ void k_copy4(const float* __restrict__ src, float* __restrict__ dst, int n4) {
  int i = blockIdx.x * blockDim.x + threadIdx.x;
  if (i < n4) ((float4*)dst)[i] = ((const float4*)src)[i];
}

__global__ void k_deg(const int* __restrict__ ei, const float* __restrict__ wt,
                      float* __restrict__ deg) {
  int e = blockIdx.x * blockDim.x + threadIdx.x;
  if (e < NEDGES) atomicAdd(&deg[ei[NEDGES + e]], wt[e]);
}

__global__ void k_dis(const float* __restrict__ deg, float* __restrict__ dis) {
  int i = blockIdx.x * blockDim.x + threadIdx.x;
  if (i < NNODES) {
    float d = deg[i];
    dis[i] = (d > 0.0f) ? rsqrtf(d) : 0.0f;
  }
}

// one wave (32 lanes) per edge; each lane handles 4 channels (float4 gather)
__global__ void k_agg(const int* __restrict__ ei, const float* __restrict__ wt,
                      const float* __restrict__ dis, const float* __restrict__ x,
                      float* __restrict__ out) {
  int gid  = blockIdx.x * blockDim.x + threadIdx.x;
  int e    = gid >> 5;
  int lane = gid & 31;
  if (e >= NEDGES) return;
  int r = ei[e];
  int c = ei[NEDGES + e];
  float w = dis[r] * wt[e] * dis[c];
  const float4 xv = *(const float4*)(x + (size_t)r * DF + lane * 4);
  float* o = out + (size_t)c * DF + lane * 4;
  atomicAdd(o + 0, w * xv.x);
  atomicAdd(o + 1, w * xv.y);
  atomicAdd(o + 2, w * xv.z);
  atomicAdd(o + 3, w * xv.w);
}

// per-channel sum / sumsq over NNODES rows (128 channels)
__global__ void k_bnstats(const float* __restrict__ h, float* __restrict__ sum,
                          float* __restrict__ sumsq) {
  int c  = threadIdx.x & (DF - 1);
  int rg = threadIdx.x >> 7;  // 0/1 with blockDim 256
  float s = 0.0f, q = 0.0f;
  for (int r = blockIdx.x * 2 + rg; r < NNODES; r += gridDim.x * 2) {
    float v = h[(size_t)r * DF + c];
    s += v;
    q += v * v;
  }
  __shared__ float ls[256], lq[256];
  ls[threadIdx.x] = s;
  lq[threadIdx.x] = q;
  __syncthreads();
  if (rg == 0) {
    atomicAdd(&sum[c],   ls[c] + ls[DF + c]);
    atomicAdd(&sumsq[c], lq[c] + lq[DF + c]);
  }
}

__global__ void k_bnfin(const float* __restrict__ sum, const float* __restrict__ sumsq,
                        float* __restrict__ mu, float* __restrict__ rstd) {
  int c = threadIdx.x;
  if (c < DF) {
    const float inv = 1.0f / (float)NNODES;
    float m = sum[c] * inv;
    float v = sumsq[c] * inv - m * m;  // biased variance
    if (v < 0.0f) v = 0.0f;
    mu[c]   = m;
    rstd[c] = rsqrtf(v + BN_EPS);
  }
}

__global__ void k_bnrelu(const float* __restrict__ t, const float* __restrict__ mu,
                         const float* __restrict__ rstd, const float* __restrict__ g,
                         const float* __restrict__ b, float* __restrict__ h) {
  int i = blockIdx.x * blockDim.x + threadIdx.x;
  if (i < NNODES * DF) {
    int c = i & (DF - 1);
    float v = (t[i] - mu[c]) * rstd[c] * g[c] + b[c];
    h[i] = v > 0.0f ? v : 0.0f;
  }
}

// W[Nc][128] row-major -> pair-packed transpose:
//   Wp[(k>>1) * 2*Nc + 2*n + (k&1)] = W[n][k]
// so the (k, k+1) B-operand pair for column n is one aligned 8-byte load and
// consecutive lanes (n, n+1, ...) read consecutive 8-byte cells (coalesced).
__global__ void k_packW(const float* __restrict__ W, float* __restrict__ Wp, int Nc) {
  int i = blockIdx.x * blockDim.x + threadIdx.x;
  if (i < Nc * DF) {
    int n = i / DF, k = i % DF;
    Wp[(k >> 1) * (2 * Nc) + 2 * n + (k & 1)] = W[i];
  }
}

// --------------------------- WMMA f32 GEMM kernel --------------------------
// D[M x NC] = preop(A)[M x 128] @ W^T (+ bias) (+= existing D + bias if ACCUM)
// PREOP: 0 = identity, 1 = BN+ReLU (mu,rstd,g,bb), 2 = ReLU
// One wave per (M-tile, N-tile): 25000 waves for NC=128.

template <int PREOP, bool ACCUM, int NC>
__global__ void k_gemm(const float* __restrict__ A, const float* __restrict__ Wp,
                       const float* __restrict__ bias, const float* __restrict__ mu,
                       const float* __restrict__ rstd, const float* __restrict__ g,
                       const float* __restrict__ bb, float* __restrict__ D) {
  constexpr int MT = NNODES / 16;  // 3125 (exact)
  constexpr int NB = NC / 16;
  constexpr int NW = MT * NB;
  int w = blockIdx.x * (blockDim.x >> 5) + (threadIdx.x >> 5);
  if (w >= NW) return;  // wave-uniform: EXEC stays all-ones for WMMA
  const int nb    = w % NB;
  const int mtile = w / NB;

  const int lane = threadIdx.x & 31;
  const int half = lane >> 4;   // 0: K pair {4k,4k+1} / M 0..7 ; 1: {4k+2,4k+3} / M 8..15
  const int l16  = lane & 15;
  const int n    = nb * 16 + l16;

  // --- preload B tile: one aligned b64 per k-step, coalesced across lanes ---
  v2f breg[32];
#pragma unroll
  for (int kk = 0; kk < 32; ++kk) {
    breg[kk] = *(const v2f*)(Wp + (size_t)(2 * kk + half) * (2 * NC) + 2 * n);
  }

  // --- preload + pre-op A tile (16x128) into registers (32 x v2f) ---
  const float* arow = A + ((size_t)mtile * 16 + l16) * DF;
  v2f areg[32];
#pragma unroll
  for (int kk = 0; kk < 32; ++kk) {
    const int c0 = 4 * kk + 2 * half;
    v2f a = *(const v2f*)(arow + c0);
    if constexpr (PREOP == 1) {
      v2f m = *(const v2f*)(mu + c0);
      v2f r = *(const v2f*)(rstd + c0);
      v2f gg = *(const v2f*)(g + c0);
      v2f bv = *(const v2f*)(bb + c0);
      a.x = (a.x - m.x) * r.x * gg.x + bv.x;
      a.y = (a.y - m.y) * r.y * gg.y + bv.y;
    }
    if constexpr (PREOP >= 1) {
      a.x = fmaxf(a.x, 0.0f);
      a.y = fmaxf(a.y, 0.0f);
    }
    areg[kk] = a;
  }

  // --- two interleaved accumulator chains ---
  v8f acc0, acc1;
  const float bv = bias[n];
  if constexpr (ACCUM) {
#pragma unroll
    for (int i = 0; i < 8; ++i) {
      acc0[i] = D[((size_t)mtile * 16 + half * 8 + i) * NC + n] + bv;
      acc1[i] = 0.0f;
    }
  } else {
#pragma unroll
    for (int i = 0; i < 8; ++i) {
      acc0[i] = bv;
      acc1[i] = 0.0f;
    }
  }

#pragma unroll
  for (int kk = 0; kk < 32; kk += 2) {
    acc0 = __builtin_amdgcn_wmma_f32_16x16x4_f32(false, areg[kk],     false, breg[kk],
                                                 (short)0, acc0, false, false);
    acc1 = __builtin_amdgcn_wmma_f32_16x16x4_f32(false, areg[kk + 1], false, breg[kk + 1],
                                                 (short)0, acc1, false, false);
  }

#pragma unroll
  for (int i = 0; i < 8; ++i)
    D[((size_t)mtile * 16 + half * 8 + i) * NC + n] = acc0[i] + acc1[i];
}

// ------------------------------- host driver -------------------------------

extern "C" void kernel_launch(void* const* d_in, const int* in_sizes, int n_in,
                              void* d_out, int out_size, void* d_ws, size_t ws_size,
                              hipStream_t stream) {
  // setup_inputs() order
  const float* x     = (const float*)d_in[0];
  const int*   ei    = (const int*)d_in[1];   // [2, E]; row = ei[0:E], col = ei[E:2E]
  const float* wt    = (const float*)d_in[2];
  const float* c0W1  = (const float*)d_in[3];
  const float* c0b1  = (const float*)d_in[4];
  const float* c0bg  = (const float*)d_in[5];
  const float* c0bb  = (const float*)d_in[6];
  const float* c0W2  = (const float*)d_in[7];
  const float* c0b2  = (const float*)d_in[8];
  const float* ob0g  = (const float*)d_in[9];
  const float* ob0b  = (const float*)d_in[10];
  const float* c1W1  = (const float*)d_in[11];
  const float* c1b1  = (const float*)d_in[12];
  const float* c1bg  = (const float*)d_in[13];
  const float* c1bb  = (const float*)d_in[14];
  const float* c1W2  = (const float*)d_in[15];
  const float* c1b2  = (const float*)d_in[16];
  const float* ob1g  = (const float*)d_in[17];
  const float* ob1b  = (const float*)d_in[18];
  const float* lp0W  = (const float*)d_in[19];
  const float* lp0b  = (const float*)d_in[20];
  const float* lp2W  = (const float*)d_in[21];
  const float* lp2b  = (const float*)d_in[22];
  const float* outW  = (const float*)d_in[23];
  const float* outb  = (const float*)d_in[24];

  // workspace layout (floats); all sub-buffers 8-byte aligned
  float* ws    = (float*)d_ws;
  float* bufA  = ws;                                   // N*128
  float* bufB  = bufA + (size_t)NNODES * DF;           // N*128
  float* deg   = bufB + (size_t)NNODES * DF;           // N
  float* dis   = deg + NNODES;                         // N
  float* sum   = dis + NNODES;                         // 128
  float* sumsq = sum + DF;                             // 128
  float* mu    = sumsq + DF;                           // 128
  float* rstd  = mu + DF;                              // 128
  float* Wp    = rstd + DF;                            // 128*128 packed-weight slot

  const int NF  = NNODES * DF;
  const dim3 B256(256);
  auto gz = [](int n) { return dim3((n + 255) / 256); };

  // ---- degree / normalization coefficients ----
  k_zero<<<gz(NNODES), B256, 0, stream>>>(deg, NNODES);
  k_deg<<<gz(NEDGES), B256, 0, stream>>>(ei, wt, deg);
  k_dis<<<gz(NNODES), B256, 0, stream>>>(deg, dis);

  const dim3 GB(256);                                   // 8 waves / block
  const dim3 GG128((NNODES / 16) * (DF / 16) / 8);      // 25000 waves -> 3125 blocks
  const dim3 GG64(((NNODES / 16) * (DOUT / 16) + 7) / 8);
  const int  EG = (NEDGES * 32 + 255) / 256;            // wave-per-edge aggregation

  auto stats = [&](const float* t) {
    k_zero<<<gz(DF), B256, 0, stream>>>(sum, DF);
    k_zero<<<gz(DF), B256, 0, stream>>>(sumsq, DF);
    k_bnstats<<<128, B256, 0, stream>>>(t, sum, sumsq);
    k_bnfin<<<1, 128, 0, stream>>>(sum, sumsq, mu, rstd);
  };
  auto xpose = [&](const float* W, int Nc) {
    k_packW<<<gz(Nc * DF), B256, 0, stream>>>(W, Wp, Nc);
  };

  // ================= layer 0 =================
  k_copy4<<<gz(NF / 4), B256, 0, stream>>>(x, bufA, NF / 4);   // agg = x + ...
  k_agg<<<EG, B256, 0, stream>>>(ei, wt, dis, x, bufA);

  xpose(c0W1, DF);
  k_gemm<0, false, DF><<<GG128, GB, 0, stream>>>(bufA, Wp, c0b1, nullptr, nullptr,
                                                 nullptr, nullptr, bufB);   // t1
  stats(bufB);
  xpose(c0W2, DF);
  k_gemm<1, false, DF><<<GG128, GB, 0, stream>>>(bufB, Wp, c0b2, mu, rstd,
                                                 c0bg, c0bb, bufA);         // t2
  stats(bufA);
  k_bnrelu<<<gz(NF), B256, 0, stream>>>(bufA, mu, rstd, ob0g, ob0b, bufB);  // h0

  // ================= layer 1 =================
  k_copy4<<<gz(NF / 4), B256, 0, stream>>>(bufB, bufA, NF / 4);
  k_agg<<<EG, B256, 0, stream>>>(ei, wt, dis, bufB, bufA);

  xpose(c1W1, DF);
  k_gemm<0, false, DF><<<GG128, GB, 0, stream>>>(bufA, Wp, c1b1, nullptr, nullptr,
                                                 nullptr, nullptr, bufB);   // t1'
  stats(bufB);
  xpose(c1W2, DF);
  k_gemm<1, false, DF><<<GG128, GB, 0, stream>>>(bufB, Wp, c1b2, mu, rstd,
                                                 c1bg, c1bb, bufA);         // t2'
  stats(bufA);
  k_bnrelu<<<gz(NF), B256, 0, stream>>>(bufA, mu, rstd, ob1g, ob1b, bufB);  // h1

  // ================= prediction heads =================
  xpose(lp0W, DF);
  k_gemm<0, false, DF><<<GG128, GB, 0, stream>>>(x, Wp, lp0b, nullptr, nullptr,
                                                 nullptr, nullptr, bufA);   // t = x@lp0
  xpose(lp2W, DF);
  k_gemm<0, true, DF><<<GG128, GB, 0, stream>>>(bufB, Wp, lp2b, nullptr, nullptr,
                                                nullptr, nullptr, bufA);    // t += h1@lp2
  xpose(outW, DOUT);
  k_gemm<2, false, DOUT><<<GG64, GB, 0, stream>>>(bufA, Wp, outb, nullptr, nullptr,
                                                  nullptr, nullptr, (float*)d_out);
}